// ContrastiveLoss_3813930958991
// MI455X (gfx1250) — compile-verified
//
#include <hip/hip_runtime.h>
#include <math.h>
#include <stdint.h>

typedef float v2f __attribute__((ext_vector_type(2)));
typedef float v8f __attribute__((ext_vector_type(8)));

#define B_N 4096
#define D_N 512
#define C_N 1000
#define TM 64   /* macro-tile rows */
#define TN 128  /* macro-tile cols */
#define KC 32   /* k-chunk */
#define LDSTR (KC + 4) /* 36 words: 36*m mod 64 distinct for m=0..15 */

// ---------------------------------------------------------------- utilities
__global__ void init_counts_kernel(int* __restrict__ counts) {
    for (int i = threadIdx.x; i < 1024; i += blockDim.x) counts[i] = 0;
}

__global__ void count_labels_kernel(const int* __restrict__ labels,
                                    int* __restrict__ counts) {
    int idx = blockIdx.x * blockDim.x + threadIdx.x;
    if (idx < B_N) atomicAdd(&counts[labels[idx]], 1); // int atomics: deterministic
}

// u_i = x_i / ||x_i||  (so cos(i,j) = u_i . u_j)
__global__ __launch_bounds__(128) void normalize_rows_kernel(
        const float* __restrict__ xs, float* __restrict__ U) {
    const int row = blockIdx.x;
    const int tid = threadIdx.x; // 128 threads, 4 floats each
    const float4* x4 = (const float4*)(xs + (size_t)row * D_N);
    float4 a = x4[tid];
    float ss = a.x * a.x + a.y * a.y + a.z * a.z + a.w * a.w;
    __shared__ float red[128];
    red[tid] = ss;
    __syncthreads();
    for (int s = 64; s > 0; s >>= 1) {
        if (tid < s) red[tid] += red[tid + s];
        __syncthreads();
    }
    float n = sqrtf(red[0]);
    float inv = (n > 1e-12f) ? 1.0f / n : 0.0f;
    a.x *= inv; a.y *= inv; a.z *= inv; a.w *= inv;
    ((float4*)(U + (size_t)row * D_N))[tid] = a;
}

// ------------------------------------------------- loss1: row cross-entropy
__global__ __launch_bounds__(256) void ce_rows_kernel(
        const float* __restrict__ yb, const int* __restrict__ gth,
        float* __restrict__ out_rows) {
    const int row = blockIdx.x;
    const int tid = threadIdx.x;
    const float* r = yb + (size_t)row * C_N;
    __shared__ float red[256];
    float mx = -INFINITY;
    for (int c = tid; c < C_N; c += 256) mx = fmaxf(mx, r[c]);
    red[tid] = mx;
    __syncthreads();
    for (int s = 128; s > 0; s >>= 1) {
        if (tid < s) red[tid] = fmaxf(red[tid], red[tid + s]);
        __syncthreads();
    }
    mx = red[0];
    __syncthreads();
    float se = 0.f;
    for (int c = tid; c < C_N; c += 256) se += expf(r[c] - mx);
    red[tid] = se;
    __syncthreads();
    for (int s = 128; s > 0; s >>= 1) {
        if (tid < s) red[tid] += red[tid + s];
        __syncthreads();
    }
    if (tid == 0) {
        int t = gth[row];
        out_rows[row] = -(r[t] - mx - logf(red[0]));
    }
}

// ---------------- loss2: fused WMMA Gram tile + same-label/triu reduction
// Async global->LDS (ASYNCcnt) double buffering; each wave register-blocks a
// 32x32 output (2x2 WMMA tiles) so 4 LDS fragment loads feed 4 WMMAs.
__device__ __forceinline__ void async_b128_to_lds(uint32_t lds_addr,
                                                  const float* gptr) {
    asm volatile("global_load_async_to_lds_b128 %0, %1, off"
                 :: "v"(lds_addr), "v"(gptr)
                 : "memory");
}

__global__ __launch_bounds__(256) void gram_pos_kernel(
        const float* __restrict__ U, const int* __restrict__ labels,
        float* __restrict__ psum, int* __restrict__ pcnt) {
    __shared__ float lA[2][TM * LDSTR]; // 18 KB
    __shared__ float lB[2][TN * LDSTR]; // 36 KB
    __shared__ int labA[TM];
    __shared__ int labB[TN];
    __shared__ float redf[256];
    __shared__ int redi[256];

    const int tid = threadIdx.x;
    const int row0 = blockIdx.y * TM;
    const int col0 = blockIdx.x * TN;
    const int bidx = blockIdx.y * gridDim.x + blockIdx.x;

    if (row0 >= col0 + TN) { // whole tile strictly below diagonal: no i<j pairs
        if (tid == 0) { psum[bidx] = 0.0f; pcnt[bidx] = 0; }
        return;
    }
    if (tid < TM) labA[tid] = labels[row0 + tid];
    if (tid < TN) labB[tid] = labels[col0 + tid];

    const int wid = tid >> 5;
    const int lane = tid & 31;
    const int wr2 = wid >> 2; // 0..1 -> 32-row block
    const int wc2 = wid & 3;  // 0..3 -> 32-col block
    const int half = lane >> 4;
    const int mrow = lane & 15;
    // staging: 8 threads cover one 32-float row (float4 each); 32 rows/pass
    const int rg = tid >> 3;         // 0..31 row group
    const int colq = (tid & 7) * 4;  // column (floats)

    const uint32_t lA0 = (uint32_t)(uintptr_t)(&lA[0][0]); // low 32 bits of a
    const uint32_t lA1 = (uint32_t)(uintptr_t)(&lA[1][0]); // flat LDS pointer
    const uint32_t lB0 = (uint32_t)(uintptr_t)(&lB[0][0]); // == LDS byte offset
    const uint32_t lB1 = (uint32_t)(uintptr_t)(&lB[1][0]);

    v8f acc00 = {0.f,0.f,0.f,0.f,0.f,0.f,0.f,0.f};
    v8f acc01 = acc00, acc10 = acc00, acc11 = acc00;

    // 6 async instructions per wave per chunk: 2 (A tile) + 4 (B tile)
#define STAGE_CHUNK(LA, LB, KB)                                               \
    do {                                                                      \
        _Pragma("unroll")                                                     \
        for (int r = rg; r < TM; r += 32)                                     \
            async_b128_to_lds((LA) + (uint32_t)((r * LDSTR + colq) * 4),      \
                              U + (size_t)(row0 + r) * D_N + (KB) + colq);    \
        _Pragma("unroll")                                                     \
        for (int r = rg; r < TN; r += 32)                                     \
            async_b128_to_lds((LB) + (uint32_t)((r * LDSTR + colq) * 4),      \
                              U + (size_t)(col0 + r) * D_N + (KB) + colq);    \
    } while (0)

    // f32 16x16x4 fragment layout: lane<16 holds {K,K+1}, lane>=16 {K+2,K+3}
    // of row/col (lane&15) -> identical addressing for A and B^T tiles.
#define COMPUTE_CHUNK(BUF)                                                    \
    do {                                                                      \
        const float* pa0 = &lA[BUF][(wr2 * 32 + mrow) * LDSTR + 2 * half];    \
        const float* pa1 = pa0 + 16 * LDSTR;                                  \
        const float* pb0 = &lB[BUF][(wc2 * 32 + mrow) * LDSTR + 2 * half];    \
        const float* pb1 = pb0 + 16 * LDSTR;                                  \
        _Pragma("unroll")                                                     \
        for (int k = 0; k < KC; k += 4) {                                     \
            v2f a0 = *(const v2f*)(pa0 + k);                                  \
            v2f a1 = *(const v2f*)(pa1 + k);                                  \
            v2f b0 = *(const v2f*)(pb0 + k);                                  \
            v2f b1 = *(const v2f*)(pb1 + k);                                  \
            acc00 = __builtin_amdgcn_wmma_f32_16x16x4_f32(                    \
                false, a0, false, b0, (short)0, acc00, false, false);         \
            acc01 = __builtin_amdgcn_wmma_f32_16x16x4_f32(                    \
                false, a0, false, b1, (short)0, acc01, false, false);         \
            acc10 = __builtin_amdgcn_wmma_f32_16x16x4_f32(                    \
                false, a1, false, b0, (short)0, acc10, false, false);         \
            acc11 = __builtin_amdgcn_wmma_f32_16x16x4_f32(                    \
                false, a1, false, b1, (short)0, acc11, false, false);         \
        }                                                                     \
    } while (0)

    STAGE_CHUNK(lA0, lB0, 0); // prologue: chunk 0 -> buffer 0

    int buf = 0;
    for (int kb = 0; kb < D_N - KC; kb += KC, buf ^= 1) {
        // queue next chunk into the other buffer (safe: last compute on it
        // finished before the trailing barrier of the previous iteration)
        STAGE_CHUNK(buf ? lA0 : lA1, buf ? lB0 : lB1, kb + KC);
        // async loads complete in order: <=6 outstanding => chunk kb done
        asm volatile("s_wait_asynccnt 0x6" ::: "memory");
        __syncthreads();
        COMPUTE_CHUNK(buf);
        __syncthreads();
    }
    asm volatile("s_wait_asynccnt 0x0" ::: "memory"); // epilogue chunk
    __syncthreads();
    COMPUTE_CHUNK(buf);
#undef STAGE_CHUNK
#undef COMPUTE_CHUNK

    // C/D layout: lanes 0-15 -> M=v, lanes 16-31 -> M=v+8; N = lane&15
    float ps = 0.f;
    int pc = 0;
    {
        const v8f* accs[2][2] = {{&acc00, &acc01}, {&acc10, &acc11}};
#pragma unroll
        for (int si = 0; si < 2; ++si) {
#pragma unroll
            for (int sj = 0; sj < 2; ++sj) {
                const v8f a = *accs[si][sj];
                const int nIdx = wc2 * 32 + sj * 16 + mrow;
                const int gj = col0 + nIdx;
                const int labj = labB[nIdx];
#pragma unroll
                for (int v = 0; v < 8; ++v) {
                    const int mIdx = wr2 * 32 + si * 16 + half * 8 + v;
                    const int gi = row0 + mIdx;
                    if (gi < gj && labA[mIdx] == labj) {
                        ps += 1.0f - a[v];
                        pc += 1;
                    }
                }
            }
        }
    }
    redf[tid] = ps;
    redi[tid] = pc;
    __syncthreads();
    for (int s = 128; s > 0; s >>= 1) {
        if (tid < s) { redf[tid] += redf[tid + s]; redi[tid] += redi[tid + s]; }
        __syncthreads();
    }
    if (tid == 0) { psum[bidx] = redf[0]; pcnt[bidx] = redi[0]; }
}

// -------- loss3: first n_neg differing-label pairs in row-major triu order
__global__ __launch_bounds__(512) void loss3_kernel(
        const float* __restrict__ U, const int* __restrict__ labels,
        const int* __restrict__ counts, const int* __restrict__ n_negative,
        float* __restrict__ out) {
    const int NT = 512;
    const int tid = threadIdx.x;
    __shared__ int redi[512];
    __shared__ float redf[512];
    __shared__ float rowU[512];
    __shared__ int taken;

    int nd = 0;
    for (int c = tid; c < C_N; c += NT) nd += (counts[c] > 1) ? 1 : 0;
    redi[tid] = nd;
    __syncthreads();
    for (int s = 256; s > 0; s >>= 1) {
        if (tid < s) redi[tid] += redi[tid + s];
        __syncthreads();
    }
    const int n_neg = max(n_negative[0], redi[0]);
    if (tid == 0) taken = 0;
    __syncthreads();

    float acc = 0.f;
    for (int i = 0; i < B_N - 1; ++i) {
        if (taken >= n_neg) break; // uniform
        const int labi = labels[i];
        rowU[tid] = U[(size_t)i * D_N + tid];
        __syncthreads();
        for (int j0 = i + 1; j0 < B_N; j0 += NT) {
            const int j = j0 + tid;
            const int flag = (j < B_N && labels[j] != labi) ? 1 : 0;
            redi[tid] = flag;
            __syncthreads();
            for (int off = 1; off < NT; off <<= 1) { // inclusive scan (ranks)
                int t = (tid >= off) ? redi[tid - off] : 0;
                __syncthreads();
                redi[tid] += t;
                __syncthreads();
            }
            if (flag && (taken + redi[tid]) <= n_neg) {
                const float* uj = U + (size_t)j * D_N;
                float d = 0.f;
                for (int k = 0; k < D_N; ++k) d = fmaf(rowU[k], uj[k], d);
                acc += fmaxf(d, 0.f);
            }
            __syncthreads();
            if (tid == 0) taken += redi[NT - 1];
            __syncthreads();
            if (taken >= n_neg) break; // uniform
        }
        __syncthreads();
    }
    redf[tid] = acc;
    __syncthreads();
    for (int s = 256; s > 0; s >>= 1) {
        if (tid < s) redf[tid] += redf[tid + s];
        __syncthreads();
    }
    if (tid == 0) out[0] = redf[0] / (float)n_neg;
}

// ------------------------------------------- final deterministic combine
__global__ __launch_bounds__(1024) void finalize_kernel(
        const float* __restrict__ l1rows, const float* __restrict__ psum,
        const int* __restrict__ pcnt, const float* __restrict__ l3,
        float* __restrict__ out) {
    const int NT = 1024;
    const int tid = threadIdx.x;
    __shared__ float r1[1024];
    __shared__ float r2[1024];
    __shared__ int r3[1024];
    float s1 = 0.f;
    for (int i = tid; i < B_N; i += NT) s1 += l1rows[i];
    float s2 = 0.f;
    int c2 = 0;
    const int NB = (B_N / TM) * (B_N / TN);
    for (int i = tid; i < NB; i += NT) { s2 += psum[i]; c2 += pcnt[i]; }
    r1[tid] = s1; r2[tid] = s2; r3[tid] = c2;
    __syncthreads();
    for (int s = 512; s > 0; s >>= 1) {
        if (tid < s) { r1[tid] += r1[tid + s]; r2[tid] += r2[tid + s]; r3[tid] += r3[tid + s]; }
        __syncthreads();
    }
    if (tid == 0) {
        float loss1 = r1[0] / (float)B_N;
        float loss2 = (r3[0] > 0) ? (r2[0] / (float)r3[0]) : 0.f;
        out[0] = loss1 + loss2 + l3[0];
    }
}

extern "C" void kernel_launch(void* const* d_in, const int* in_sizes, int n_in,
                              void* d_out, int out_size, void* d_ws, size_t ws_size,
                              hipStream_t stream) {
    (void)in_sizes; (void)n_in; (void)out_size; (void)ws_size;
    const float* xs = (const float*)d_in[0];
    const float* ybars = (const float*)d_in[1];
    const int* labels = (const int*)d_in[2];
    const int* nneg = (const int*)d_in[3];

    char* ws = (char*)d_ws;
    float* U      = (float*)(ws);                       // 4096*512*4 = 8 MB
    int*   counts = (int*)(ws + 8388608);               // 4 KB (1024 ints)
    float* psum   = (float*)(ws + 8392704);             // 2048 floats
    int*   pcnt   = (int*)(ws + 8425472);               // 2048 ints
    float* l1rows = (float*)(ws + 8458240);             // 4096 floats
    float* l3     = (float*)(ws + 8474624);             // 1 float

    init_counts_kernel<<<1, 256, 0, stream>>>(counts);
    count_labels_kernel<<<B_N / 256, 256, 0, stream>>>(labels, counts);
    normalize_rows_kernel<<<B_N, 128, 0, stream>>>(xs, U);
    gram_pos_kernel<<<dim3(B_N / TN, B_N / TM), 256, 0, stream>>>(U, labels, psum, pcnt);
    ce_rows_kernel<<<B_N, 256, 0, stream>>>(ybars, labels, l1rows);
    loss3_kernel<<<1, 512, 0, stream>>>(U, labels, counts, nneg, l3);
    finalize_kernel<<<1, 1024, 0, stream>>>(l1rows, psum, pcnt, l3, (float*)d_out);
}